// MultiHeadAttentionQuantum_65481071400379
// MI455X (gfx1250) — compile-verified
//
#include <hip/hip_runtime.h>

typedef __attribute__((ext_vector_type(2))) float v2f;
typedef __attribute__((ext_vector_type(4))) float v4f;
typedef __attribute__((ext_vector_type(8))) float v8f;

// Padded LDS row stride (floats). bank = (4*row + col) % 64 -> conflict-free
// for every access pattern used below, and keeps b64/b128 alignment legal.
#define LDST 68

__global__ __launch_bounds__(128) void quantum_proj_fused(
    const float* __restrict__ x,      // [262144, 64] row-major
    const float* __restrict__ W_in,   // [64 wires, 64 embed] row-major
    const float* __restrict__ theta,  // [64]
    const float* __restrict__ W_out,  // [64 out, 64 wires] row-major
    float* __restrict__ out)          // [262144, 64]
{
    __shared__ float sWin [64 * LDST];
    __shared__ float sWout[64 * LDST];
    __shared__ float sQ[4][16 * LDST];   // one 16x64 q tile per wave

    const int tid   = threadIdx.x;
    const int wave  = tid >> 5;
    const int lane  = tid & 31;
    const int nrow  = lane & 15;          // M (A row) or N (B/C col) index
    const int klo   = (lane >> 4) << 1;   // lanes 0-15 hold K=0,1 ; 16-31 hold K=2,3
    const int mbase = (lane >> 4) << 3;   // C/D: lanes 16-31 hold rows M+8

    // ---- stage both weight matrices into padded LDS (coalesced b128) ----
    #pragma unroll
    for (int i = 0; i < 8; ++i) {
        int idx = (i * 128 + tid) * 4;    // flat element index, 4-aligned, no row crossing
        int row = idx >> 6;
        int col = idx & 63;
        *(v4f*)&sWin [row * LDST + col] = *(const v4f*)&W_in [idx];
        *(v4f*)&sWout[row * LDST + col] = *(const v4f*)&W_out[idx];
    }
    __syncthreads();

    // per-lane theta: wire index in C/D layout depends only on lane
    float th[4];
    #pragma unroll
    for (int nt = 0; nt < 4; ++nt) th[nt] = theta[nt * 16 + nrow];

    const long tile = (long)blockIdx.x * 4 + wave;  // 16-row tile per wave
    const long row0 = tile * 16;

    // ---- A fragments of x, loaded straight from global in 16x4 A layout ----
    v2f a[16];
    const float* xrow = x + (row0 + nrow) * 64 + klo;
    #pragma unroll
    for (int kk = 0; kk < 16; ++kk)
        a[kk] = *(const v2f*)(xrow + kk * 4);       // global_load_b64

    // ---- GEMM1 (token = x @ W_in^T), fused cos, stage q to LDS ----
    float* q = sQ[wave];
    #pragma unroll
    for (int nt = 0; nt < 4; ++nt) {
        v8f acc = {};
        #pragma unroll
        for (int kk = 0; kk < 16; ++kk) {
            // B[e,w] = W_in[w,e]: lane -> column w, klo -> K rows (consecutive e in memory)
            v2f b = *(const v2f*)&sWin[(nt * 16 + nrow) * LDST + kk * 4 + klo];
            acc = __builtin_amdgcn_wmma_f32_16x16x4_f32(
                      false, a[kk], false, b, (short)0, acc, false, false);
        }
        #pragma unroll
        for (int r = 0; r < 8; ++r) {
            float t = acc[r] + th[nt];
            // v_cos_f32 computes cos(2*pi*x): pre-scale by 1/(2*pi) (inline constant)
            float c = __builtin_amdgcn_cosf(t * 0.15915494309189535f);
            q[(mbase + r) * LDST + nt * 16 + nrow] = c;   // conflict-free ds_store_b32
        }
    }

    // ---- reload q in A layout (wave-private LDS: DS ops in-order, no barrier) ----
    v2f a2[16];
    #pragma unroll
    for (int kk = 0; kk < 16; ++kk)
        a2[kk] = *(const v2f*)&q[nrow * LDST + kk * 4 + klo];  // ds_load_b64

    // ---- GEMM2 (out = q @ W_out^T) and store ----
    float* obase = out + row0 * 64;
    #pragma unroll
    for (int nt = 0; nt < 4; ++nt) {
        v8f acc = {};
        #pragma unroll
        for (int kk = 0; kk < 16; ++kk) {
            v2f b = *(const v2f*)&sWout[(nt * 16 + nrow) * LDST + kk * 4 + klo];
            acc = __builtin_amdgcn_wmma_f32_16x16x4_f32(
                      false, a2[kk], false, b, (short)0, acc, false, false);
        }
        #pragma unroll
        for (int r = 0; r < 8; ++r)  // lanes 0-15 + 16-31 each write a contiguous 64B row segment
            obase[(mbase + r) * 64 + nt * 16 + nrow] = acc[r];
    }
}

extern "C" void kernel_launch(void* const* d_in, const int* in_sizes, int n_in,
                              void* d_out, int out_size, void* d_ws, size_t ws_size,
                              hipStream_t stream) {
    const float* x     = (const float*)d_in[0];
    const float* W_in  = (const float*)d_in[1];
    const float* theta = (const float*)d_in[2];
    const float* W_out = (const float*)d_in[3];
    float* out = (float*)d_out;

    const int rows   = in_sizes[0] / 64;   // 32*8192 = 262144
    const int blocks = rows / 64;          // 4 waves * 16 rows per block

    hipLaunchKernelGGL(quantum_proj_fused, dim3(blocks), dim3(128), 0, stream,
                       x, W_in, theta, W_out, out);
}